// Rasterize_18820546691597
// MI455X (gfx1250) — compile-verified
//
#include <hip/hip_runtime.h>
#include <hip/hip_bf16.h>

typedef __attribute__((ext_vector_type(2))) float v2f;
typedef __attribute__((ext_vector_type(8))) float v8f;

#define IS    192
#define NF    512
#define NB    2
#define TT    4
#define NEARP 0.1f
#define FARP  100.0f
#define EPSI  1e-3f

// One wave rasterizes a strip of 16 pixels against all 512 faces using
// V_WMMA_F32_16X16X4_F32:  D[16 rows = 4 faces x {n0,n1,n2,q}][16 pixels]
//   n_k = cofactor-row_k . (xp, yp, 1)   (edge funcs == barycentric numerators)
//   q   = sum_k n_k * (1/z_k)            (=> zp = det / q)
// Inner loop is division-free: depth compares are done by cross-multiplication
// (all accepted faces have det > 0 and q > 0), one divide per pixel at the end.
__global__ __launch_bounds__(256) void rast_wmma_kernel(
    const float* __restrict__ faces,     // [B, F, 3, 3]
    const float* __restrict__ tex,       // [B, F, 4,4,4, 3]
    float* __restrict__ rgb_out,         // [B, IS, IS, 3]
    float* __restrict__ alpha_out,       // [B, 1, IS, IS]
    float* __restrict__ depth_out)       // [B, IS, IS]
{
    // A-operand table, laid out per-chunk exactly as the WMMA A operand wants:
    // chunk c, lane l: a0 = ldsA[(c*32+l)*2], a1 = ldsA[(c*32+l)*2+1]
    __shared__ float ldsA[(NF / 4) * 64];   // 8192 floats = 32 KB
    __shared__ float ldsDet[NF];            // sign-preserving clamped determinant
    __shared__ float ldsIZ[NF * 3];         // 1/z per vertex

    const int b   = blockIdx.y;
    const int tid = threadIdx.x;

    // ---------------- Phase 1: per-face coefficient setup ----------------
    for (int f = tid; f < NF; f += 256) {
        const float* fp = faces + (size_t)(b * NF + f) * 9;
        float x0 = fp[0], y0 = fp[1], z0 = fp[2];
        float x1 = fp[3], y1 = fp[4], z1 = fp[5];
        float x2 = fp[6], y2 = fp[7], z2 = fp[8];

        // cofactor rows: n_k = rk_x * xp + rk_y * yp + rk_c
        float r0x = y1 - y2, r0y = x2 - x1, r0c = x1 * y2 - x2 * y1;
        float r1x = y2 - y0, r1y = x0 - x2, r1c = x2 * y0 - x0 * y2;
        float r2x = y0 - y1, r2y = x1 - x0, r2c = x0 * y1 - x1 * y0;
        float det = x2 * (y0 - y1) + x0 * (y1 - y2) + x1 * (y2 - y0);
        // Sign-preserving clamp: front test becomes (detc > 0), division uses
        // the same value. Tiny-negative dets are back-facing (culled) either way.
        float detc = (fabsf(det) < 1e-12f) ? ((det >= 0.0f) ? 1e-12f : -1e-12f)
                                           : det;
        float iz0 = 1.0f / z0, iz1 = 1.0f / z1, iz2 = 1.0f / z2;
        // q row: sum_k row_k * invz_k  (still linear in (xp,yp,1))
        float qx = r0x * iz0 + r1x * iz1 + r2x * iz2;
        float qy = r0y * iz0 + r1y * iz1 + r2y * iz2;
        float qc = r0c * iz0 + r1c * iz1 + r2c * iz2;

        float cx[4] = { r0x, r1x, r2x, qx };
        float cy[4] = { r0y, r1y, r2y, qy };
        float cc[4] = { r0c, r1c, r2c, qc };

        const int c  = f >> 2;
        const int fl = f & 3;
        float* base = &ldsA[c * 64];
        #pragma unroll
        for (int j = 0; j < 4; ++j) {
            int row = fl * 4 + j;              // M row in A (0..15)
            base[row * 2 + 0]        = cx[j];  // lane row      : K0 = cx
            base[row * 2 + 1]        = cy[j];  //               : K1 = cy
            base[(row + 16) * 2 + 0] = cc[j];  // lane row+16   : K2 = c1
            base[(row + 16) * 2 + 1] = 0.0f;   //               : K3 = 0
        }
        ldsDet[f] = detc;
        ldsIZ[f * 3 + 0] = iz0;
        ldsIZ[f * 3 + 1] = iz1;
        ldsIZ[f * 3 + 2] = iz2;
    }
    __syncthreads();

    // ---------------- Phase 2: WMMA raster loop ----------------
    const int lane = tid & 31;
    const int wv   = tid >> 5;
    const int tile = blockIdx.x * 8 + wv;   // 16-pixel strip index
    const int n    = lane & 15;             // pixel within strip (= N column)
    const int p    = tile * 16 + n;         // linear pixel id
    const int prow = p / IS, pcol = p % IS;
    const float xn = (2.0f * (float)pcol + 1.0f - (float)IS) * (1.0f / (float)IS);
    const float yn = (2.0f * (float)prow + 1.0f - (float)IS) * (1.0f / (float)IS);

    // B operand (4x16): lanes 0-15 -> (K0,K1) = (xp,yp); lanes 16-31 -> (K2,K3) = (1,0)
    v2f bmat;
    bmat.x = (lane < 16) ? xn : 1.0f;
    bmat.y = (lane < 16) ? yn : 0.0f;

    // depth tracked as a fraction bestNum/bestDen (both > 0 for accepted faces)
    float bestNum = FARP;
    float bestDen = 1.0f;
    int   bestF   = NF;
    const int fsel = (lane >> 4) << 1;      // lanes 0-15: faces {0,1}; 16-31: {2,3}

    for (int c = 0; c < NF / 4; ++c) {
        v2f a = *(const v2f*)&ldsA[(c * 32 + lane) * 2];
        v8f zero = {};
        v8f acc = __builtin_amdgcn_wmma_f32_16x16x4_f32(
            false, a, false, bmat, (short)0, zero, false, false);

        const int f0 = c * 4 + fsel;
        // ---- face f0 : rows 0..3 of this lane's half ----
        {
            float n0 = acc[0], n1 = acc[1], n2 = acc[2], q = acc[3];
            float det = ldsDet[f0];
            bool ok = (det > 0.0f) & (n0 >= 0.0f) & (n1 >= 0.0f) & (n2 >= 0.0f)
                      & (det > NEARP * q)            // zp > NEAR
                      & (det < FARP * q)             // zp < FAR
                      & (det * bestDen < bestNum * q); // zp < best
            bestNum = ok ? det : bestNum;
            bestDen = ok ? q   : bestDen;
            bestF   = ok ? f0  : bestF;
        }
        // ---- face f0+1 : rows 4..7 ----
        {
            float n0 = acc[4], n1 = acc[5], n2 = acc[6], q = acc[7];
            float det = ldsDet[f0 + 1];
            bool ok = (det > 0.0f) & (n0 >= 0.0f) & (n1 >= 0.0f) & (n2 >= 0.0f)
                      & (det > NEARP * q)
                      & (det < FARP * q)
                      & (det * bestDen < bestNum * q);
            bestNum = ok ? det      : bestNum;
            bestDen = ok ? q        : bestDen;
            bestF   = ok ? (f0 + 1) : bestF;
        }
    }

    // merge the two half-wave candidates for this pixel (tie -> lowest face id)
    {
        float oN = __shfl_xor(bestNum, 16, 32);
        float oQ = __shfl_xor(bestDen, 16, 32);
        int   oF = __shfl_xor(bestF, 16, 32);
        float lhs = oN * bestDen, rhs = bestNum * oQ;
        bool take = (lhs < rhs) || ((lhs == rhs) && (oF < bestF));
        bestNum = take ? oN : bestNum;
        bestDen = take ? oQ : bestDen;
        bestF   = take ? oF : bestF;
    }
    const float bestD = bestNum / bestDen;  // == FARP when nothing hit

    // ---------------- Phase 3: shade winning face (lanes 0..15) ----------------
    if (lane < 16) {
        const bool found = bestD < FARP;
        float r = 0.0f, g = 0.0f, bcol = 0.0f;
        if (found) {
            const int f  = bestF;
            const int c  = f >> 2;
            const int fl = f & 3;
            const float* basec = &ldsA[c * 64];
            float detc = ldsDet[f];          // already clamped, > 0 for winners

            float w[3];
            float s = 0.0f;
            #pragma unroll
            for (int j = 0; j < 3; ++j) {
                int row = fl * 4 + j;
                float nv = basec[row * 2 + 0] * xn + basec[row * 2 + 1] * yn
                         + basec[(row + 16) * 2 + 0];
                float wj = nv / detc;
                wj = fminf(fmaxf(wj, 0.0f), 1.0f);
                w[j] = wj;
                s += wj;
            }
            float inv_s = 1.0f / fmaxf(s, 1e-12f);

            int   lo[3];
            float fr[3];
            #pragma unroll
            for (int j = 0; j < 3; ++j) {
                float t = w[j] * inv_s * (float)(TT - 1) * bestD * ldsIZ[f * 3 + j];
                t = fminf(fmaxf(t, 0.0f), (float)(TT - 1) - EPSI);
                float tf = floorf(t);
                lo[j] = (int)tf;
                fr[j] = t - tf;
            }

            const float* tp = tex + (size_t)(b * NF + f) * (TT * TT * TT * 3);
            #pragma unroll
            for (int cc = 0; cc < 8; ++cc) {
                int b0 = cc & 1, b1 = (cc >> 1) & 1, b2 = (cc >> 2) & 1;
                float cw = (b0 ? fr[0] : 1.0f - fr[0])
                         * (b1 ? fr[1] : 1.0f - fr[1])
                         * (b2 ? fr[2] : 1.0f - fr[2]);
                int i0 = lo[0] + b0, i1 = lo[1] + b1, i2 = lo[2] + b2;
                const float* tx = tp + ((i0 * TT + i1) * TT + i2) * 3;
                r    += cw * tx[0];
                g    += cw * tx[1];
                bcol += cw * tx[2];
            }
        }
        const size_t pi = (size_t)b * (IS * IS) + (size_t)p;
        rgb_out[pi * 3 + 0] = r;
        rgb_out[pi * 3 + 1] = g;
        rgb_out[pi * 3 + 2] = bcol;
        alpha_out[pi] = found ? 1.0f : 0.0f;
        depth_out[pi] = bestD;   // FAR when nothing hit, matching reference
    }
}

extern "C" void kernel_launch(void* const* d_in, const int* in_sizes, int n_in,
                              void* d_out, int out_size, void* d_ws, size_t ws_size,
                              hipStream_t stream) {
    const float* faces = (const float*)d_in[0];   // [2, 512, 3, 3]
    const float* tex   = (const float*)d_in[1];   // [2, 512, 4,4,4, 3]

    float* out   = (float*)d_out;
    float* rgb   = out;                                         // B*IS*IS*3
    float* alpha = out + (size_t)NB * IS * IS * 3;              // B*IS*IS
    float* depth = alpha + (size_t)NB * IS * IS;                // B*IS*IS

    // 8 waves per block, each wave owns a 16-pixel strip: 36864/128 = 288 blocks per batch
    dim3 grid(IS * IS / (16 * 8), NB);
    rast_wmma_kernel<<<grid, dim3(256), 0, stream>>>(faces, tex, rgb, alpha, depth);
}